// PredictLayer_48816598286774
// MI455X (gfx1250) — compile-verified
//
#include <hip/hip_runtime.h>

// ---------------------------------------------------------------------------
// Shapes: B=16, D=64, N=200, T=48, H=8, DK=8
// ---------------------------------------------------------------------------
#define BB 16
#define DD 64
#define NN 200
#define TT 48

typedef __attribute__((ext_vector_type(2))) float v2f;
typedef __attribute__((ext_vector_type(8))) float v8f;

__device__ __forceinline__ v8f wmma_f32_k4(v2f a, v2f b, v8f c) {
  // V_WMMA_F32_16X16X4_F32 : D = A(16x4) * B(4x16) + C(16x16), fp32 throughout
  return __builtin_amdgcn_wmma_f32_16x16x4_f32(false, a, false, b, (short)0, c,
                                               false, false);
}

// ---------------------------------------------------------------------------
// Kernel 0: transpose weights into (K x N) row-major B-matrix layout in ws.
// ws layout (floats):
//   [0,12288)      WqT[j][c][dout]
//   [12288,24576)  WkT[j][c][dout]
//   [24576,28672)  W1T[c][dout]
//   [28672,32768)  W2T[c][dout]
//   [32768,36864)  ThetaT[c][dout]
//   [36864, +B*N*T*D) h buffer
// ---------------------------------------------------------------------------
__global__ void stgnn_wtrans(const float* __restrict__ Wq,
                             const float* __restrict__ Wk,
                             const float* __restrict__ W1,
                             const float* __restrict__ W2,
                             const float* __restrict__ Theta,
                             float* __restrict__ ws) {
  int i = blockIdx.x * blockDim.x + threadIdx.x;
  if (i < 12288) {
    int j = i / 4096;
    int r = i - j * 4096;
    int c = r >> 6;
    int dout = r & 63;
    ws[i]         = Wq[(dout * 64 + c) * 3 + j];
    ws[12288 + i] = Wk[(dout * 64 + c) * 3 + j];
  }
  if (i < 4096) {
    int c = i >> 6;
    int dout = i & 63;
    ws[24576 + i] = W1[dout * 64 + c];
    ws[28672 + i] = W2[dout * 64 + c];
    ws[32768 + i] = Theta[dout * 64 + c];
  }
}

// ---------------------------------------------------------------------------
// Kernel 1: temporal attention + LN + residual -> h   (one block per (b,n))
// Dynamic LDS layout (floats):
//   xT   [0,3072)    48x64
//   Qs   [3072)      48x64
//   Ks   [6144)      48x64
//   Vs   [9216)      48x64
//   Os   [12288)     48x64
//   TOs  [15360)     48x65
//   mean [18480) 48 ; rstd [18528) 48
//   S    [18576)     4 waves x (48x49)
// total 27984 floats = 111936 B
// ---------------------------------------------------------------------------
__global__ __launch_bounds__(128) void stgnn_temporal(
    const float* __restrict__ x, const float* __restrict__ wsw,
    const float* __restrict__ bq, const float* __restrict__ bk,
    const float* __restrict__ b1, const float* __restrict__ b2,
    const float* __restrict__ g0, const float* __restrict__ beta0,
    float* __restrict__ hout_g) {
  extern __shared__ float sm[];
  float* xT   = sm;
  float* Qs   = sm + 3072;
  float* Ks   = sm + 6144;
  float* Vs   = sm + 9216;
  float* Os   = sm + 12288;
  float* TOs  = sm + 15360;
  float* mean = sm + 18480;
  float* rstd = sm + 18528;
  float* Sall = sm + 18576;

  const int tid = threadIdx.x;
  const int lane = tid & 31;
  const int wave = tid >> 5;
  const int bn = blockIdx.x;
  const int b = bn / NN;
  const int n = bn - b * NN;

  const float* WqT = wsw;
  const float* WkT = wsw + 12288;
  const float* W1T = wsw + 24576;
  const float* W2T = wsw + 28672;

  // stage x[b,:,n,:] transposed -> xT[t][d]
  const float* xb = x + (size_t)b * (DD * NN * TT) + (size_t)n * TT;
  for (int i = tid; i < TT * DD; i += 128) {
    int d = i / TT;
    int t = i - d * TT;
    xT[t * DD + d] = xb[(size_t)d * (NN * TT) + t];
  }
  __syncthreads();

  // WMMA lane->element mapping (wave32, f32 16x16x4)
  const int mr = lane & 15;         // A row / C col group
  const int kk = (lane >> 4) << 1;  // K pair selector
  const int nc = lane & 15;         // B/C column
  const int rb = (lane >> 4) * 8;   // C row base

  // ---- Q,K (1x3 conv over T) and V (linear) : 12 output tiles of 16x16 ----
  for (int tt = wave; tt < 12; tt += 4) {
    const int m0 = (tt >> 2) * 16;
    const int n0 = (tt & 3) * 16;
    v8f qa = {};
    v8f ka = {};
    v8f va = {};
    for (int j = 0; j < 3; ++j) {
      const float* Bq = WqT + j * 4096;
      const float* Bk = WkT + j * 4096;
      const int shift = j - 1;
      for (int k = 0; k < DD; k += 4) {
        const int row = m0 + mr + shift;
        v2f a;
        if (row >= 0 && row < TT) {
          a.x = xT[row * DD + k + kk];
          a.y = xT[row * DD + k + kk + 1];
        } else {
          a.x = 0.f;
          a.y = 0.f;
        }
        v2f bb;
        bb.x = Bq[(k + kk) * DD + n0 + nc];
        bb.y = Bq[(k + kk + 1) * DD + n0 + nc];
        qa = wmma_f32_k4(a, bb, qa);
        bb.x = Bk[(k + kk) * DD + n0 + nc];
        bb.y = Bk[(k + kk + 1) * DD + n0 + nc];
        ka = wmma_f32_k4(a, bb, ka);
      }
    }
    for (int k = 0; k < DD; k += 4) {
      v2f a;
      a.x = xT[(m0 + mr) * DD + k + kk];
      a.y = xT[(m0 + mr) * DD + k + kk + 1];
      v2f bb;
      bb.x = W1T[(k + kk) * DD + n0 + nc];
      bb.y = W1T[(k + kk + 1) * DD + n0 + nc];
      va = wmma_f32_k4(a, bb, va);
    }
    const float biq = bq[n0 + nc];
    const float bik = bk[n0 + nc];
    const float biv = b1[n0 + nc];
#pragma unroll
    for (int r = 0; r < 8; ++r) {
      Qs[(m0 + rb + r) * DD + n0 + nc] = qa[r] + biq;
      Ks[(m0 + rb + r) * DD + n0 + nc] = ka[r] + bik;
      Vs[(m0 + rb + r) * DD + n0 + nc] = va[r] + biv;
    }
  }
  __syncthreads();

  // ---- per-head attention: scores (K=8), softmax, P@V (K=48) ----
  float* Sw = Sall + wave * (TT * 49);
  for (int hh = wave; hh < 8; hh += 4) {
    const int cb = hh * 8;
    for (int mi = 0; mi < 3; ++mi) {
      for (int si = 0; si < 3; ++si) {
        v8f acc = {};
        for (int k = 0; k < 8; k += 4) {
          v2f a;
          a.x = Qs[(mi * 16 + mr) * DD + cb + k + kk];
          a.y = Qs[(mi * 16 + mr) * DD + cb + k + kk + 1];
          v2f bb;  // B[k][s] = K[s][cb+k]
          bb.x = Ks[(si * 16 + nc) * DD + cb + k + kk];
          bb.y = Ks[(si * 16 + nc) * DD + cb + k + kk + 1];
          acc = wmma_f32_k4(a, bb, acc);
        }
#pragma unroll
        for (int r = 0; r < 8; ++r)
          Sw[(mi * 16 + rb + r) * 49 + si * 16 + nc] = acc[r] * 0.35355339f;
      }
    }
    // row softmax (stride 49 -> conflict-free lane-per-row)
    for (int row = lane; row < TT; row += 32) {
      float mx = -1e30f;
      for (int s = 0; s < TT; ++s) mx = fmaxf(mx, Sw[row * 49 + s]);
      float sum = 0.f;
      for (int s = 0; s < TT; ++s) {
        float e = __expf(Sw[row * 49 + s] - mx);
        Sw[row * 49 + s] = e;
        sum += e;
      }
      float inv = 1.f / sum;
      for (int s = 0; s < TT; ++s) Sw[row * 49 + s] *= inv;
    }
    // O_h = P @ V_h  (48x8; compute 16-wide, store left 8 cols)
    for (int mi = 0; mi < 3; ++mi) {
      v8f acc = {};
      for (int k = 0; k < TT; k += 4) {
        v2f a;
        a.x = Sw[(mi * 16 + mr) * 49 + k + kk];
        a.y = Sw[(mi * 16 + mr) * 49 + k + kk + 1];
        v2f bb;
        if (nc < 8) {
          bb.x = Vs[(k + kk) * DD + cb + nc];
          bb.y = Vs[(k + kk + 1) * DD + cb + nc];
        } else {
          bb.x = 0.f;
          bb.y = 0.f;
        }
        acc = wmma_f32_k4(a, bb, acc);
      }
      if (nc < 8) {
#pragma unroll
        for (int r = 0; r < 8; ++r)
          Os[(mi * 16 + rb + r) * DD + cb + nc] = acc[r];
      }
    }
  }
  __syncthreads();

  // ---- t_out = O @ W2^T + b2 ----
  for (int tt = wave; tt < 12; tt += 4) {
    const int m0 = (tt >> 2) * 16;
    const int n0 = (tt & 3) * 16;
    v8f acc = {};
    for (int k = 0; k < DD; k += 4) {
      v2f a;
      a.x = Os[(m0 + mr) * DD + k + kk];
      a.y = Os[(m0 + mr) * DD + k + kk + 1];
      v2f bb;
      bb.x = W2T[(k + kk) * DD + n0 + nc];
      bb.y = W2T[(k + kk + 1) * DD + n0 + nc];
      acc = wmma_f32_k4(a, bb, acc);
    }
    const float bias = b2[n0 + nc];
#pragma unroll
    for (int r = 0; r < 8; ++r)
      TOs[(m0 + rb + r) * 65 + n0 + nc] = acc[r] + bias;
  }
  __syncthreads();

  // ---- LayerNorm(t_out) + residual -> h ----
  for (int t = tid; t < TT; t += 128) {
    float s = 0.f, s2 = 0.f;
    for (int d = 0; d < DD; ++d) {
      float v = TOs[t * 65 + d];
      s += v;
      s2 += v * v;
    }
    float m = s * (1.f / DD);
    mean[t] = m;
    rstd[t] = rsqrtf(s2 * (1.f / DD) - m * m + 1e-5f);
  }
  __syncthreads();
  float* hg = hout_g + (size_t)bn * (TT * DD);
  for (int i = tid; i < TT * DD; i += 128) {
    int t = i >> 6;
    int d = i & 63;
    hg[i] = xT[t * DD + d] +
            (TOs[t * 65 + d] - mean[t]) * rstd[t] * g0[d] + beta0[d];
  }
}

// ---------------------------------------------------------------------------
// Kernel 2: spatial GCN + LN + residual -> out   (one block per (b,t))
// Dynamic LDS layout (floats):
//   Xs  [0)      200x65
//   Ag  [13000)  200x65
//   S   [26000)  4 waves x (16x201)
//   R   [38864)  4 waves x (16x66)
//   Rm  [43088) 64 ; Rr [43152) 64
// total 43216 floats = 172864 B
// ---------------------------------------------------------------------------
__global__ __launch_bounds__(128) void stgnn_spatial(
    const float* __restrict__ hbuf, const float* __restrict__ adj,
    const float* __restrict__ ThetaT, const float* __restrict__ g1,
    const float* __restrict__ beta1, float* __restrict__ out) {
  extern __shared__ float sm[];
  float* Xs = sm;
  float* Ag = sm + 13000;
  float* Sb = sm + 26000;
  float* Rb = sm + 38864;
  float* Rm = sm + 43088;
  float* Rr = sm + 43152;

  const int tid = threadIdx.x;
  const int lane = tid & 31;
  const int wave = tid >> 5;
  const int bt = blockIdx.x;
  const int b = bt / TT;
  const int t = bt - b * TT;

  // stage X = h[b,:,t,:]  (200 x 64, padded stride 65)
  for (int i = tid; i < NN * DD; i += 128) {
    int n = i >> 6;
    int d = i & 63;
    Xs[n * 65 + d] = hbuf[(((size_t)b * NN + n) * TT + t) * DD + d];
  }
  __syncthreads();

  const int mr = lane & 15;
  const int kk = (lane >> 4) << 1;
  const int nc = lane & 15;
  const int rb = (lane >> 4) * 8;

  // ---- per 16-row block: scores, softmax*adj, aggregation ----
  float* Sw = Sb + wave * (16 * 201);
  for (int blk = wave; blk < 13; blk += 4) {
    const int m0 = blk * 16;
    // scores S = X X^T / 8   (13 col-tiles, K=64)
    for (int si = 0; si < 13; ++si) {
      const int n0 = si * 16;
      v8f acc = {};
      for (int k = 0; k < DD; k += 4) {
        v2f a;
        const int row = m0 + mr;
        if (row < NN) {
          a.x = Xs[row * 65 + k + kk];
          a.y = Xs[row * 65 + k + kk + 1];
        } else {
          a.x = 0.f;
          a.y = 0.f;
        }
        v2f bb;  // B[k][n] = X[n][k]
        const int col = n0 + nc;
        if (col < NN) {
          bb.x = Xs[col * 65 + k + kk];
          bb.y = Xs[col * 65 + k + kk + 1];
        } else {
          bb.x = 0.f;
          bb.y = 0.f;
        }
        acc = wmma_f32_k4(a, bb, acc);
      }
#pragma unroll
      for (int r = 0; r < 8; ++r)
        Sw[(rb + r) * 201 + n0 + nc] = acc[r] * 0.125f;
    }
    // softmax over 200 cols, then * adj * (1/sqrt(D))
    if (lane < 16) {
      const int row = m0 + lane;
      if (row < NN) {
        float mx = -1e30f;
        for (int c = 0; c < NN; ++c) mx = fmaxf(mx, Sw[lane * 201 + c]);
        float sum = 0.f;
        for (int c = 0; c < NN; ++c) {
          float e = __expf(Sw[lane * 201 + c] - mx);
          Sw[lane * 201 + c] = e;
          sum += e;
        }
        const float sc = 0.125f / sum;
        for (int c = 0; c < NN; ++c)
          Sw[lane * 201 + c] *= sc * adj[row * NN + c];
      }
    }
    // agg = P @ X  (16 x 64, K=200 -> exactly 50 WMMA steps)
    for (int ni = 0; ni < 4; ++ni) {
      v8f acc = {};
      for (int k = 0; k < NN; k += 4) {
        v2f a;
        a.x = Sw[mr * 201 + k + kk];
        a.y = Sw[mr * 201 + k + kk + 1];
        v2f bb;
        bb.x = Xs[(k + kk) * 65 + ni * 16 + nc];
        bb.y = Xs[(k + kk + 1) * 65 + ni * 16 + nc];
        acc = wmma_f32_k4(a, bb, acc);
      }
#pragma unroll
      for (int r = 0; r < 8; ++r) {
        const int row = m0 + rb + r;
        if (row < NN) Ag[row * 65 + ni * 16 + nc] = acc[r];
      }
    }
  }
  __syncthreads();

  // ---- s_out = relu(Ag @ Theta^T); LN; residual; store ----
  float* Rw = Rb + wave * (16 * 66);
  for (int mt = wave; mt < 13; mt += 4) {
    const int m0 = mt * 16;
    for (int ni = 0; ni < 4; ++ni) {
      v8f acc = {};
      for (int k = 0; k < DD; k += 4) {
        v2f a;
        const int row = m0 + mr;
        if (row < NN) {
          a.x = Ag[row * 65 + k + kk];
          a.y = Ag[row * 65 + k + kk + 1];
        } else {
          a.x = 0.f;
          a.y = 0.f;
        }
        v2f bb;
        bb.x = ThetaT[(k + kk) * DD + ni * 16 + nc];
        bb.y = ThetaT[(k + kk + 1) * DD + ni * 16 + nc];
        acc = wmma_f32_k4(a, bb, acc);
      }
#pragma unroll
      for (int r = 0; r < 8; ++r)
        Rw[(rb + r) * 66 + ni * 16 + nc] = fmaxf(acc[r], 0.f);
    }
    if (lane < 16) {
      float s = 0.f, s2 = 0.f;
      for (int d = 0; d < DD; ++d) {
        float v = Rw[lane * 66 + d];
        s += v;
        s2 += v * v;
      }
      float m = s * (1.f / DD);
      Rm[wave * 16 + lane] = m;
      Rr[wave * 16 + lane] = rsqrtf(s2 * (1.f / DD) - m * m + 1e-5f);
    }
    for (int i = lane; i < 16 * DD; i += 32) {
      const int r = i >> 6;
      const int d = i & 63;
      const int row = m0 + r;
      if (row < NN) {
        float v = (Rw[r * 66 + d] - Rm[wave * 16 + r]) * Rr[wave * 16 + r] *
                      g1[d] +
                  beta1[d];
        out[(((size_t)b * NN + row) * TT + t) * DD + d] = Xs[row * 65 + d] + v;
      }
    }
  }
}

// ---------------------------------------------------------------------------
extern "C" void kernel_launch(void* const* d_in, const int* in_sizes, int n_in,
                              void* d_out, int out_size, void* d_ws,
                              size_t ws_size, hipStream_t stream) {
  (void)in_sizes;
  (void)n_in;
  (void)out_size;
  (void)ws_size;
  const float* x     = (const float*)d_in[0];
  const float* adj   = (const float*)d_in[1];
  const float* Wq    = (const float*)d_in[2];
  const float* bq    = (const float*)d_in[3];
  const float* Wk    = (const float*)d_in[4];
  const float* bk    = (const float*)d_in[5];
  const float* W1    = (const float*)d_in[6];
  const float* b1    = (const float*)d_in[7];
  const float* W2    = (const float*)d_in[8];
  const float* b2    = (const float*)d_in[9];
  const float* Theta = (const float*)d_in[10];
  const float* g0    = (const float*)d_in[11];
  const float* beta0 = (const float*)d_in[12];
  const float* g1    = (const float*)d_in[13];
  const float* beta1 = (const float*)d_in[14];
  float* ws = (float*)d_ws;
  float* out = (float*)d_out;
  float* hbuf = ws + 36864;

  stgnn_wtrans<<<48, 256, 0, stream>>>(Wq, Wk, W1, W2, Theta, ws);

  const size_t smA = 27984u * sizeof(float);  // 112 KB
  stgnn_temporal<<<BB * NN, 128, smA, stream>>>(x, ws, bq, bk, b1, b2, g0,
                                                beta0, hbuf);

  const size_t smB = 43216u * sizeof(float);  // 173 KB
  stgnn_spatial<<<BB * TT, 128, smB, stream>>>(hbuf, adj, ws + 32768, g1,
                                               beta1, out);
}